// XModel_53609781788737
// MI455X (gfx1250) — compile-verified
//
#include <hip/hip_runtime.h>
#include <hip/hip_bf16.h>

typedef __attribute__((ext_vector_type(2))) float v2f;
typedef __attribute__((ext_vector_type(8))) float v8f;

#define NROWS   131072      // 64 * 2048
#define T       2048
#define F       1024
#define BATCH32 32
#define KSEL    20
#define FEAT_BLOCK 655360   // 32*20*1024

// ---------------------------------------------------------------------------
// Kernel 1: masked squared magnitudes via V_WMMA_F32_16X16X4_F32.
// One wave handles 16 tokens. Per iteration each lane loads 8B of the 16x4
// fp32 tile; the same registers serve as A (16x4) and B (4x16 = A^T), so the
// 16x16 f32 accumulator's diagonal is sum-of-squares per token (exact fp32).
// ---------------------------------------------------------------------------
__global__ void mag2_kernel(const float* __restrict__ features,
                            const float* __restrict__ mask_nor,
                            const float* __restrict__ mask_abn,
                            float* __restrict__ mag2out) {
    const int lane  = threadIdx.x & 31;
    const int wv    = threadIdx.x >> 5;
    const long tile = (long)blockIdx.x * 8 + wv;     // 8192 tiles of 16 tokens
    const long row0 = tile * 16;

    const int m     = lane & 15;   // token within tile
    const int khalf = lane >> 4;   // lanes 16-31 hold K=2,3
    const float* p  = features + (row0 + m) * (long)F + khalf * 2;

    v8f acc0 = {};
    v8f acc1 = {};
#pragma unroll 8
    for (int c = 0; c < F; c += 8) {
        v2f a0 = *(const v2f*)(p + c);
        v2f a1 = *(const v2f*)(p + c + 4);
        // D = A * A^T + C  (same regs as A and B); fp32 in, fp32 accumulate
        acc0 = __builtin_amdgcn_wmma_f32_16x16x4_f32(
            false, a0, false, a0, (short)0, acc0, false, false);
        acc1 = __builtin_amdgcn_wmma_f32_16x16x4_f32(
            false, a1, false, a1, (short)0, acc1, false, false);
    }
    v8f acc = acc0 + acc1;

    // Diagonal (M==N): lanes 0-7 -> tokens 0-7 (VGPR=lane), lanes 24-31 ->
    // tokens 8-15 (VGPR=lane-24). VGPR index is lane&7 in both cases.
    const int sel = lane & 7;
    float d = acc[0];
    if (sel == 1) d = acc[1];
    if (sel == 2) d = acc[2];
    if (sel == 3) d = acc[3];
    if (sel == 4) d = acc[4];
    if (sel == 5) d = acc[5];
    if (sel == 6) d = acc[6];
    if (sel == 7) d = acc[7];

    const bool valid = (lane < 8) || (lane >= 24);
    if (valid) {
        const int  mt  = (lane < 8) ? lane : (lane - 16);
        const long row = row0 + mt;
        const int  b   = (int)(row >> 11);      // /2048
        const int  t   = (int)(row & (T - 1));  // %2048
        const float mu = (b < BATCH32) ? mask_nor[b * T + t]
                                       : mask_abn[(b - BATCH32) * T + t];
        // rank on mag^2 with 0 for dropped entries: same ordering (and same
        // tie/edge behavior) as reference's mag * drop
        mag2out[row] = (mu > 0.1f) ? d : 0.0f;
    }
}

// ---------------------------------------------------------------------------
// Kernel 2: top-20 per row (64 rows x 2048). Iterative argmax in LDS,
// min-index tie-break == jax.lax.top_k ordering.
// ---------------------------------------------------------------------------
__global__ void topk_kernel(const float* __restrict__ mag2,
                            int* __restrict__ idxout) {
    __shared__ float vals[T];
    __shared__ float redv[8];
    __shared__ int   redi[8];

    const int r    = blockIdx.x;
    const int tid  = threadIdx.x;
    const int lane = tid & 31;
    const int wv   = tid >> 5;

    for (int i = tid; i < T; i += 256) vals[i] = mag2[r * T + i];
    __syncthreads();

    for (int it = 0; it < KSEL; ++it) {
        float bv = -1.0f;
        int   bi = 0x7fffffff;
        for (int i = tid; i < T; i += 256) {
            float v = vals[i];
            if (v > bv) { bv = v; bi = i; }
        }
        // wave32 reduction
        for (int off = 16; off > 0; off >>= 1) {
            float ov = __shfl_xor(bv, off, 32);
            int   oi = __shfl_xor(bi, off, 32);
            if (ov > bv || (ov == bv && oi < bi)) { bv = ov; bi = oi; }
        }
        if (lane == 0) { redv[wv] = bv; redi[wv] = bi; }
        __syncthreads();
        if (tid == 0) {
            float BV = redv[0]; int BI = redi[0];
            for (int w = 1; w < 8; ++w) {
                if (redv[w] > BV || (redv[w] == BV && redi[w] < BI)) {
                    BV = redv[w]; BI = redi[w];
                }
            }
            idxout[r * KSEL + it] = BI;
            vals[BI] = -1.0f;   // exclude for next round
        }
        __syncthreads();
    }
}

// ---------------------------------------------------------------------------
// Kernel 3: gather selected feature rows (4KB each) to output, float4 copies.
// Output layout: [score_abn(32)][score_nor(32)][abn_feat 32*20*1024][nor_feat]
// ---------------------------------------------------------------------------
__global__ void gather_kernel(const float* __restrict__ features,
                              const int* __restrict__ idx,
                              float* __restrict__ out) {
    const int flat = blockIdx.x;       // 64*20 blocks
    const int r    = flat / KSEL;
    const int k    = flat % KSEL;
    const int tok  = idx[r * KSEL + k];

    const float4* src =
        (const float4*)(features + ((long)r * T + tok) * (long)F);

    long base;
    if (r >= BATCH32)   // abnormal half
        base = 64 + ((long)(r - BATCH32) * KSEL + k) * (long)F;
    else                // normal half
        base = 64 + FEAT_BLOCK + ((long)r * KSEL + k) * (long)F;

    float4* dst = (float4*)(out + base);
    dst[threadIdx.x] = src[threadIdx.x];   // 256 threads * 16B = 4KB
}

// ---------------------------------------------------------------------------
// Kernel 4: mean of the 20 selected scores per row.
// ---------------------------------------------------------------------------
__global__ void score_kernel(const float* __restrict__ scores,
                             const int* __restrict__ idx,
                             float* __restrict__ out) {
    const int r = threadIdx.x;   // 64 threads
    float s = 0.0f;
    for (int k = 0; k < KSEL; ++k)
        s += scores[(long)r * T + idx[r * KSEL + k]];
    s *= (1.0f / (float)KSEL);
    if (r >= BATCH32) out[r - BATCH32] = s;   // score_abnormal
    else              out[BATCH32 + r] = s;   // score_normal
}

extern "C" void kernel_launch(void* const* d_in, const int* in_sizes, int n_in,
                              void* d_out, int out_size, void* d_ws, size_t ws_size,
                              hipStream_t stream) {
    const float* features = (const float*)d_in[0];   // (64,2048,1024) f32
    const float* scores   = (const float*)d_in[1];   // (64,2048,1)    f32
    const float* mask_abn = (const float*)d_in[2];   // (32,2048)      f32
    const float* mask_nor = (const float*)d_in[3];   // (32,2048)      f32
    float* out = (float*)d_out;

    float* mag2 = (float*)d_ws;                        // 64*2048 floats
    int*   idx  = (int*)((char*)d_ws + NROWS * sizeof(float)); // 64*20 ints

    // 8192 tiles of 16 tokens, 8 waves (tiles) per 256-thread block
    mag2_kernel<<<NROWS / 16 / 8, 256, 0, stream>>>(features, mask_nor,
                                                    mask_abn, mag2);
    topk_kernel<<<64, 256, 0, stream>>>(mag2, idx);
    gather_kernel<<<64 * KSEL, 256, 0, stream>>>(features, idx, out);
    score_kernel<<<1, 64, 0, stream>>>(scores, idx, out);
}